// ExternalAttention_35321811043062
// MI455X (gfx1250) — compile-verified
//
#include <hip/hip_runtime.h>

typedef float v2f __attribute__((ext_vector_type(2)));
typedef float v8f __attribute__((ext_vector_type(8)));

#define BATCH 16
#define CH    512
#define NTOK  4096
#define SDIM  8
#define SPAD  16

// ---------------- Pass 1: attn logits via V_WMMA_F32_16X16X4_F32 ----------------
// logits[b][s][n] = sum_c x[b][c][n] * Wk[s][c]
// block = 128 threads (4 waves); each wave owns one 16-token tile; grid (NTOK/64, BATCH)
__global__ __launch_bounds__(128)
void ea_pass1_logits(const float* __restrict__ x, const float* __restrict__ Wk,
                     float* __restrict__ logits) {
    __shared__ float wkT[CH * SPAD];   // [c][s] zero-padded to 16 cols, 32 KB
    const int tid = threadIdx.x;
    for (int i = tid; i < CH * SPAD; i += 128) {
        const int s = i & (SPAD - 1);
        const int c = i >> 4;
        wkT[i] = (s < SDIM) ? Wk[s * CH + c] : 0.0f;
    }
    __syncthreads();

    const int b        = blockIdx.y;
    const int wave     = tid >> 5;
    const int lane     = tid & 31;
    const int laneHalf = lane >> 4;          // 0: K rows 0/1, 1: K rows 2/3 (A frag)
    const int laneMod  = lane & 15;
    const int tileBase = blockIdx.x * 64 + wave * 16;
    const int tok      = tileBase + laneMod; // A-matrix row M for this lane
    const int kOff     = laneHalf * 2;

    const float* xb    = x + (size_t)b * CH * NTOK;
    const float* bbase = wkT + (size_t)kOff * SPAD + laneMod; // B frag: row k, col laneMod

    v8f acc = {0.f, 0.f, 0.f, 0.f, 0.f, 0.f, 0.f, 0.f};

#pragma unroll 8
    for (int kk = 0; kk < CH / 4; ++kk) {
        const int c0 = kk * 4 + kOff;
        v2f a;
        a.x = xb[(size_t)c0 * NTOK + tok];           // K = c0     (coalesced over laneMod)
        a.y = xb[(size_t)(c0 + 1) * NTOK + tok];     // K = c0 + 1
        v2f bf;
        bf.x = bbase[(size_t)(kk * 4) * SPAD];       // B row kk*4   + kOff
        bf.y = bbase[(size_t)(kk * 4 + 1) * SPAD];   // B row kk*4+1 + kOff
        acc = __builtin_amdgcn_wmma_f32_16x16x4_f32(false, a, false, bf,
                                                    (short)0, acc, false, false);
    }

    // D layout: VGPR j -> (M = j + laneHalf*8, N = laneMod); N is the S index
    const int sIdx = laneMod;
    if (sIdx < SDIM) {
        float* lp = logits + ((size_t)b * SDIM + sIdx) * NTOK;
        const int tb = tileBase + laneHalf * 8;
#pragma unroll
        for (int j = 0; j < 8; ++j) lp[tb + j] = acc[j];
    }
}

// ---------------- Pass 2: per-(b,s) softmax stats over the 4096 tokens ----------------
// grid (SDIM, BATCH), 256 threads; 16 logits per thread held in registers
__global__ __launch_bounds__(256)
void ea_pass2_stats(const float* __restrict__ logits, float* __restrict__ smax,
                    float* __restrict__ ssum) {
    __shared__ float red[256];
    const int bs = blockIdx.y * SDIM + blockIdx.x;
    const float* lp = logits + (size_t)bs * NTOK;
    const int tid = threadIdx.x;

    float v[16];
    float m = -__builtin_inff();
#pragma unroll
    for (int i = 0; i < 16; ++i) {
        v[i] = lp[tid + i * 256];
        m = fmaxf(m, v[i]);
    }
    red[tid] = m;
    __syncthreads();
    for (int w = 128; w > 0; w >>= 1) {
        if (tid < w) red[tid] = fmaxf(red[tid], red[tid + w]);
        __syncthreads();
    }
    m = red[0];
    __syncthreads();

    float s = 0.f;
#pragma unroll
    for (int i = 0; i < 16; ++i) s += __expf(v[i] - m);
    red[tid] = s;
    __syncthreads();
    for (int w = 128; w > 0; w >>= 1) {
        if (tid < w) red[tid] += red[tid + w];
        __syncthreads();
    }
    if (tid == 0) { smax[bs] = m; ssum[bs] = red[0]; }
}

// ---------------- Pass 3: weights -> out = relu(x + attn @ Wv^T) ----------------
// grid (NTOK/64, BATCH), 256 threads: 64 tokens/block, 4 channels per step
__global__ __launch_bounds__(256)
void ea_pass3_out(const float* __restrict__ x, const float* __restrict__ Wv,
                  const float* __restrict__ logits, const float* __restrict__ smax,
                  const float* __restrict__ ssum, float* __restrict__ out) {
    __shared__ float wv[CH * SDIM];    // 16 KB: Wv[c][s]
    __shared__ float wgt[64 * SDIM];   // 2 KB: normalized weights per token
    const int tid = threadIdx.x;
    const int b = blockIdx.y;

    for (int i = tid; i < CH * SDIM; i += 256) wv[i] = Wv[i];

    if (tid < 64) {
        const int n = blockIdx.x * 64 + tid;
        float w[SDIM];
        float t = 0.f;
#pragma unroll
        for (int s = 0; s < SDIM; ++s) {
            const int bs = b * SDIM + s;
            const float l = logits[(size_t)bs * NTOK + n];
            const float e = __expf(l - smax[bs]) / ssum[bs];
            w[s] = e;
            t += e;
        }
        const float inv = 1.0f / (1e-9f + t);
#pragma unroll
        for (int s = 0; s < SDIM; ++s) wgt[tid * SDIM + s] = w[s] * inv;
    }
    __syncthreads();

    const int tok = tid & 63;
    const int cg  = tid >> 6;     // 0..3
    const int n   = blockIdx.x * 64 + tok;

    const float4* wrow = (const float4*)(&wgt[tok * SDIM]);
    const float4 w0 = wrow[0], w1 = wrow[1];

#pragma unroll 4
    for (int cc = 0; cc < CH / 4; ++cc) {
        const int c = cc * 4 + cg;
        const size_t idx = ((size_t)b * CH + c) * NTOK + n;
        const float4* vr = (const float4*)(&wv[c * SDIM]);
        const float4 v0 = vr[0], v1 = vr[1];
        float acc = x[idx];
        acc += w0.x * v0.x + w0.y * v0.y + w0.z * v0.z + w0.w * v0.w
             + w1.x * v1.x + w1.y * v1.y + w1.z * v1.z + w1.w * v1.w;
        out[idx] = fmaxf(acc, 0.0f);
    }
}

extern "C" void kernel_launch(void* const* d_in, const int* in_sizes, int n_in,
                              void* d_out, int out_size, void* d_ws, size_t ws_size,
                              hipStream_t stream) {
    (void)in_sizes; (void)n_in; (void)out_size; (void)ws_size;
    const float* x  = (const float*)d_in[0];
    const float* Wk = (const float*)d_in[1];
    const float* Wv = (const float*)d_in[2];
    float* out = (float*)d_out;

    float* logits = (float*)d_ws;                       // 16*8*4096 floats = 2 MB
    float* smax   = logits + (size_t)BATCH * SDIM * NTOK;
    float* ssum   = smax + BATCH * SDIM;

    ea_pass1_logits<<<dim3(NTOK / 64, BATCH), 128, 0, stream>>>(x, Wk, logits);
    ea_pass2_stats<<<dim3(SDIM, BATCH), 256, 0, stream>>>(logits, smax, ssum);
    ea_pass3_out<<<dim3(NTOK / 64, BATCH), 256, 0, stream>>>(x, Wv, logits, smax, ssum, out);
}